// HyperbolicGraphConvolutionLayer_28114855919662
// MI455X (gfx1250) — compile-verified
//
#include <hip/hip_runtime.h>
#include <hip/hip_bf16.h>

#define EPS 1e-6f
#define NN 768
#define BB 2
#define DD 32

typedef __attribute__((ext_vector_type(2))) float v2f;
typedef __attribute__((ext_vector_type(8))) float v8f;

__device__ __forceinline__ float wsum(float v) {
    for (int m = 16; m >= 1; m >>= 1) v += __shfl_xor(v, m, 32);
    return v;
}

// ---------------- Kernel A: per-node hyperbolic preprocessing ----------------
// Computes y = expmap(expmap0(proj_tan0(logmap0(x)@W_lin)), bias) and u = logmap0(y).
// One wave per node, lane = feature.
__global__ __launch_bounds__(32) void node_prep(const float* __restrict__ x,
                                                const float* __restrict__ W_lin,
                                                const float* __restrict__ b_lin,
                                                float* __restrict__ u_out,
                                                float* __restrict__ y_out) {
    int node = blockIdx.x;          // 0..B*N-1
    int l = threadIdx.x;            // 0..31
    float xv = x[(size_t)node * DD + l];
    float x0 = __shfl(xv, 0, 32);
    float dd = acoshf(fmaxf(x0, 1.0f + EPS));
    float sp = sqrtf(fmaxf(x0 * x0 - 1.0f, EPS));
    float u = (l == 0) ? 0.0f : dd / sp * xv;
    // h = proj_tan0(u @ W_lin)
    float h = 0.0f;
    for (int k = 0; k < DD; ++k) h += __shfl(u, k, 32) * W_lin[k * DD + l];
    if (l == 0) h = 0.0f;
    // y = expmap0(h)
    float n2 = wsum(h * h);
    float nn = sqrtf(fmaxf(n2, EPS));
    float y = (l == 0) ? coshf(nn) : sinhf(nn) / nn * h;
    float y0 = __shfl(y, 0, 32);
    // bias = transp0(y, proj_tan0(b_lin))
    float pb = (l == 0) ? 0.0f : b_lin[l];
    float ip = wsum((l == 0) ? 0.0f : y * pb);   // l_inner with pb0 = 0
    float coef = ip / (1.0f + y0);
    float bias = pb + coef * ((l == 0) ? (y0 + 1.0f) : y);
    // y = expmap(y, bias)
    float li = wsum((l == 0) ? -bias * bias : bias * bias);
    float un = sqrtf(fmaxf(li, EPS));
    y = coshf(un) * y + sinhf(un) / un * bias;
    // u = logmap0(y)
    y0 = __shfl(y, 0, 32);
    dd = acoshf(fmaxf(y0, 1.0f + EPS));
    sp = sqrtf(fmaxf(y0 * y0 - 1.0f, EPS));
    float u2 = (l == 0) ? 0.0f : dd / sp * y;
    u_out[(size_t)node * DD + l] = u2;
    y_out[(size_t)node * DD + l] = y;
}

// ---------------- Kernel B: P/Q/R node-linear precompute (f32 WMMA) ----------------
// P = u @ att_w1[0:32], Q = u @ att_w1[32:64], R = u @ msg_w1[0:32]
// M = 1536 node rows (96 tiles), K = 32 (8 k-steps of v_wmma_f32_16x16x4_f32), N = 32.
__global__ __launch_bounds__(32) void precompute_PQR(const float* __restrict__ u,
                                                     const float* __restrict__ att_w1,
                                                     const float* __restrict__ msg_w1,
                                                     float* __restrict__ P,
                                                     float* __restrict__ Q,
                                                     float* __restrict__ R) {
    int mt = blockIdx.x;                 // 0..95
    int L = threadIdx.x;
    int lane16 = L & 15;
    int koff = (L >= 16) ? 2 : 0;
    int row = mt * 16 + lane16;
    // A fragment: 16x4 f32 per k-step: VGPR0 = K{0|2}, VGPR1 = K{1|3}
    v2f a[8];
    for (int kk = 0; kk < 8; ++kk) {
        int k = 4 * kk + koff;
        a[kk][0] = u[(size_t)row * DD + k];
        a[kk][1] = u[(size_t)row * DD + k + 1];
    }
    const float* Ws[3] = {att_w1, att_w1 + 32 * DD, msg_w1};
    float* Os[3] = {P, Q, R};
    for (int m = 0; m < 3; ++m) {
        for (int nt = 0; nt < 2; ++nt) {
            int nc = nt * 16 + lane16;
            v8f c = {};
            for (int kk = 0; kk < 8; ++kk) {
                int k = 4 * kk + koff;
                v2f bf;
                bf[0] = Ws[m][(size_t)k * DD + nc];
                bf[1] = Ws[m][(size_t)(k + 1) * DD + nc];
                c = __builtin_amdgcn_wmma_f32_16x16x4_f32(false, a[kk], false, bf,
                                                          (short)0, c, false, false);
            }
            for (int r = 0; r < 8; ++r) {
                int mrow = mt * 16 + ((L < 16) ? r : r + 8);
                Os[m][(size_t)mrow * DD + nc] = c[r];
            }
        }
    }
}

// ---------------- Kernel C: edge kernel ----------------
// One block per (batch, i-tile of 16). 8 waves; wave w handles j-tiles w, w+8, ...
// Gram (Lorentz inner products) via f32 WMMA; per-edge att/msg scalars VALU;
// Sacc[i][n] = sum_j att_ij * relu(R_j[n] + e_ij*mw32[n] + mb1[n]); attsum[i] = sum_j att_ij.
__global__ __launch_bounds__(256) void edge_kernel(const float* __restrict__ y,
                                                   const float* __restrict__ adj,
                                                   const float* __restrict__ P,
                                                   const float* __restrict__ Q,
                                                   const float* __restrict__ R,
                                                   const float* __restrict__ att_w1,
                                                   const float* __restrict__ att_b1,
                                                   const float* __restrict__ att_w2,
                                                   const float* __restrict__ att_b2,
                                                   const float* __restrict__ msg_w1,
                                                   const float* __restrict__ msg_b1,
                                                   float* __restrict__ Sacc,
                                                   float* __restrict__ attsum_out) {
    __shared__ float att_lds[8][16][16];
    __shared__ float e_lds[8][16][16];
    __shared__ float S_lds[16][32];
    __shared__ float asum_lds[16];
    int it = blockIdx.x;             // i-tile 0..47
    int bb = blockIdx.y;             // batch
    int tid = threadIdx.x;
    int w = tid >> 5;
    int L = tid & 31;
    for (int idx = tid; idx < 16 * 32; idx += 256) S_lds[idx >> 5][idx & 31] = 0.0f;
    if (tid < 16) asum_lds[tid] = 0.0f;
    __syncthreads();

    size_t nodebase = (size_t)bb * NN;
    int i0 = it * 16;
    int lane16 = L & 15;
    int hi = (L >= 16);
    int koff = hi ? 2 : 0;

    // A fragments: y_i tile with time component (feature 0) negated -> Gram = l_inner
    v2f afr[8];
    {
        size_t rowoff = (nodebase + i0 + lane16) * DD;
        for (int kk = 0; kk < 8; ++kk) {
            int k = 4 * kk + koff;
            float a0 = y[rowoff + k];
            if (k == 0) a0 = -a0;
            afr[kk][0] = a0;
            afr[kk][1] = y[rowoff + k + 1];
        }
    }

    float S_reg[16];
    for (int q = 0; q < 16; ++q) S_reg[q] = 0.0f;
    float asum_reg = 0.0f;
    int i_loc2 = lane16;             // phase-2 row
    int nbase = hi ? 16 : 0;         // phase-2 feature half
    float b2 = att_b2[0];

    for (int jt = w; jt < 48; jt += 8) {
        int j0 = jt * 16;
        // ---- Gram tile via f32 WMMA (8 k-steps) ----
        v8f c = {};
        size_t jrow = (nodebase + j0 + lane16) * DD;
        for (int kk = 0; kk < 8; ++kk) {
            int k = 4 * kk + koff;
            v2f bf;
            bf[0] = y[jrow + k];
            bf[1] = y[jrow + k + 1];
            c = __builtin_amdgcn_wmma_f32_16x16x4_f32(false, afr[kk], false, bf,
                                                      (short)0, c, false, false);
        }
        // ---- Phase 1: per-edge scalars (dist, attention) ----
        for (int r = 0; r < 8; ++r) {
            int il = hi ? r + 8 : r;
            int gi = i0 + il, gj = j0 + lane16;
            float e = acoshf(fmaxf(-c[r], 1.0f + EPS));   // c[r] = l_inner(y_i, y_j)
            float a = adj[((size_t)bb * NN + gi) * NN + gj];
            float att = 0.0f;
            if (a != 0.0f) {
                const float* Pi = P + (nodebase + gi) * DD;
                const float* Qj = Q + (nodebase + gj) * DD;
                float t = 0.0f;
                for (int n = 0; n < DD; ++n) {
                    float z = Pi[n] + Qj[n] + a * att_w1[64 * DD + n] +
                              e * att_w1[65 * DD + n] + att_b1[n];
                    float sg = 1.0f / (1.0f + __expf(-z));   // silu(z) = z*sigmoid(z)
                    t += z * sg * att_w2[n];
                }
                att = a / (1.0f + __expf(-(t + b2)));
            }
            att_lds[w][il][lane16] = att;
            e_lds[w][il][lane16] = e;
        }
        // ---- Phase 2: weighted message accumulation (lane = (i, n-half)) ----
        for (int n2 = 0; n2 < 16; ++n2) {
            int n = nbase + n2;
            float w32 = msg_w1[32 * DD + n];
            float bm = msg_b1[n];
            float acc = 0.0f;
            for (int jl = 0; jl < 16; ++jl) {
                float av = att_lds[w][i_loc2][jl];
                if (av != 0.0f) {
                    float ev = e_lds[w][i_loc2][jl];
                    float m = R[(nodebase + j0 + jl) * DD + n] + ev * w32 + bm;
                    acc += av * fmaxf(m, 0.0f);
                }
            }
            S_reg[n2] += acc;
        }
        if (!hi) {
            for (int jl = 0; jl < 16; ++jl) asum_reg += att_lds[w][i_loc2][jl];
        }
    }
    // combine waves' partial sums in LDS
    for (int n2 = 0; n2 < 16; ++n2) atomicAdd(&S_lds[i_loc2][nbase + n2], S_reg[n2]);
    if (!hi) atomicAdd(&asum_lds[i_loc2], asum_reg);
    __syncthreads();
    for (int idx = tid; idx < 16 * 32; idx += 256) {
        int il = idx >> 5, n = idx & 31;
        Sacc[(nodebase + i0 + il) * DD + n] = S_lds[il][n];
    }
    if (tid < 16) attsum_out[nodebase + i0 + tid] = asum_lds[tid];
}

// ---------------- Kernel D: per-node finalize ----------------
// msg = Sacc @ msg_w2 + attsum*msg_b2 ; out-MLP ; transp0/expmap ; layernorm ; expmap0 chain.
__global__ __launch_bounds__(32) void finalize(const float* __restrict__ y,
                                               const float* __restrict__ Sacc,
                                               const float* __restrict__ attsum,
                                               const float* __restrict__ msg_w2,
                                               const float* __restrict__ msg_b2,
                                               const float* __restrict__ out_w1,
                                               const float* __restrict__ out_b1,
                                               const float* __restrict__ out_w2,
                                               const float* __restrict__ out_b2,
                                               const float* __restrict__ ln_g,
                                               const float* __restrict__ ln_b,
                                               float* __restrict__ out) {
    int node = blockIdx.x;
    int l = threadIdx.x;
    float s = Sacc[(size_t)node * DD + l];
    float asum = attsum[node];
    float m = 0.0f;
    for (int k = 0; k < DD; ++k) m += __shfl(s, k, 32) * msg_w2[k * DD + l];
    m += asum * msg_b2[l];
    float t = 0.0f;
    for (int k = 0; k < DD; ++k) t += __shfl(m, k, 32) * out_w1[k * DD + l];
    t = fmaxf(t + out_b1[l], 0.0f);
    float o = 0.0f;
    for (int k = 0; k < DD; ++k) o += __shfl(t, k, 32) * out_w2[k * DD + l];
    o += out_b2[l];
    // transp0(y, proj_tan0(o))
    float pm = (l == 0) ? 0.0f : o;
    float yv = y[(size_t)node * DD + l];
    float y0 = __shfl(yv, 0, 32);
    float ip = wsum((l == 0) ? 0.0f : yv * pm);
    float coef = ip / (1.0f + y0);
    float v = pm + coef * ((l == 0) ? (y0 + 1.0f) : yv);
    // h2 = expmap(y, v)
    float li = wsum((l == 0) ? -v * v : v * v);
    float un = sqrtf(fmaxf(li, EPS));
    float h2 = coshf(un) * yv + sinhf(un) / un * v;
    // u2 = logmap0(h2)
    float h20 = __shfl(h2, 0, 32);
    float dd = acoshf(fmaxf(h20, 1.0f + EPS));
    float spq = sqrtf(fmaxf(h20 * h20 - 1.0f, EPS));
    float u2 = (l == 0) ? 0.0f : dd / spq * h2;
    // layernorm over features 1..31 (31 elements)
    float mu = wsum((l == 0) ? 0.0f : u2) / 31.0f;
    float dv = (l == 0) ? 0.0f : (u2 - mu);
    float var = wsum(dv * dv) / 31.0f;
    float spn = (l == 0) ? 0.0f
                         : (u2 - mu) / sqrtf(var + 1e-5f) * ln_g[l - 1] + ln_b[l - 1];
    // h2 = expmap0([*, spn])
    float n2 = wsum((l == 0) ? 0.0f : spn * spn);
    float nn = sqrtf(fmaxf(n2, EPS));
    float h2b = (l == 0) ? coshf(nn) : sinhf(nn) / nn * spn;
    // out = expmap0(proj_tan0(relu(logmap0(h2b))))
    float g0 = __shfl(h2b, 0, 32);
    dd = acoshf(fmaxf(g0, 1.0f + EPS));
    spq = sqrtf(fmaxf(g0 * g0 - 1.0f, EPS));
    float lg = (l == 0) ? 0.0f : dd / spq * h2b;
    float rl = fmaxf(lg, 0.0f);
    float rn2 = wsum((l == 0) ? 0.0f : rl * rl);
    float rnn = sqrtf(fmaxf(rn2, EPS));
    float ov = (l == 0) ? coshf(rnn) : sinhf(rnn) / rnn * rl;
    out[(size_t)node * DD + l] = ov;
}

extern "C" void kernel_launch(void* const* d_in, const int* in_sizes, int n_in,
                              void* d_out, int out_size, void* d_ws, size_t ws_size,
                              hipStream_t stream) {
    const float* x      = (const float*)d_in[0];
    const float* adj    = (const float*)d_in[1];
    const float* W_lin  = (const float*)d_in[2];
    const float* b_lin  = (const float*)d_in[3];
    const float* att_w1 = (const float*)d_in[4];
    const float* att_b1 = (const float*)d_in[5];
    const float* att_w2 = (const float*)d_in[6];
    const float* att_b2 = (const float*)d_in[7];
    const float* msg_w1 = (const float*)d_in[8];
    const float* msg_b1 = (const float*)d_in[9];
    const float* msg_w2 = (const float*)d_in[10];
    const float* msg_b2 = (const float*)d_in[11];
    const float* out_w1 = (const float*)d_in[12];
    const float* out_b1 = (const float*)d_in[13];
    const float* out_w2 = (const float*)d_in[14];
    const float* out_b2 = (const float*)d_in[15];
    const float* ln_g   = (const float*)d_in[16];
    const float* ln_b   = (const float*)d_in[17];
    (void)in_sizes; (void)n_in; (void)out_size; (void)ws_size;

    const size_t NODES = (size_t)BB * NN;           // 1536
    const size_t NF = NODES * DD;                   // 49152
    float* ws   = (float*)d_ws;
    float* u    = ws;                               // NF
    float* yb   = ws + NF;                          // NF
    float* P    = ws + 2 * NF;                      // NF
    float* Q    = ws + 3 * NF;                      // NF
    float* R    = ws + 4 * NF;                      // NF
    float* S    = ws + 5 * NF;                      // NF
    float* asum = ws + 6 * NF;                      // NODES

    node_prep<<<(int)NODES, 32, 0, stream>>>(x, W_lin, b_lin, u, yb);
    precompute_PQR<<<(int)(NODES / 16), 32, 0, stream>>>(u, att_w1, msg_w1, P, Q, R);
    edge_kernel<<<dim3(NN / 16, BB), 256, 0, stream>>>(yb, adj, P, Q, R,
                                                       att_w1, att_b1, att_w2, att_b2,
                                                       msg_w1, msg_b1, S, asum);
    finalize<<<(int)NODES, 32, 0, stream>>>(yb, S, asum, msg_w2, msg_b2,
                                            out_w1, out_b1, out_w2, out_b2,
                                            ln_g, ln_b, (float*)d_out);
    // second tuple output: adj passthrough
    hipMemcpyAsync((float*)d_out + NF, adj, sizeof(float) * (size_t)BB * NN * NN,
                   hipMemcpyDeviceToDevice, stream);
}